// HausdorffERLoss_38912403702197
// MI455X (gfx1250) — compile-verified
//
#include <hip/hip_runtime.h>
#include <cstdint>
#include <cstddef>

// Problem geometry (fixed by the reference): (8, 1, 128,128,128) f32
#define NB        8
#define NXD       128
#define NYZ       128
#define SLICE_EL  (128*128)            // one (b,x) slice, 16384 floats
#define NSLICES   (NB*NXD)             // 1024
#define NBLK      (NSLICES*2)          // 2 half-slice blocks per slice
#define N_TOTAL   (NB*128*128*128)     // 16,777,216
#define N_BATCH   (128*128*128)        // 2,097,152

typedef float v2f  __attribute__((ext_vector_type(2)));
typedef float v4f  __attribute__((ext_vector_type(4)));
typedef float v8f  __attribute__((ext_vector_type(8)));
typedef int   i32x4 __attribute__((ext_vector_type(4)));
typedef __attribute__((address_space(1))) i32x4 gi32x4;   // global int4
typedef __attribute__((address_space(3))) i32x4 li32x4;   // LDS int4

// ---------------- workspace layout (bytes) ----------------
#define WS_TOTAL_OFF   0                         // double accumulator
#define WS_SCALE_OFF   64                        // 8 floats
#define WS_BIAS_OFF    96                        // 8 floats
#define WS_PMIN_OFF    256                       // 2048 floats
#define WS_PMAX_OFF    (256 + 8192)              // 2048 floats
#define WS_PSUM_OFF    (256 + 16384)             // 2048 doubles
#define WS_BUFA_OFF    65536
#define WS_BUFB_OFF    (65536 + (size_t)N_TOTAL*4)

// ---------------- CDNA5 async copy helpers ----------------
__device__ __forceinline__ void async_load_b128(const float* g, float* lds_generic) {
  uint32_t loff = (uint32_t)(uintptr_t)lds_generic;   // low 32 bits of generic = LDS offset
#if __has_builtin(__builtin_amdgcn_global_load_async_to_lds_b128)
  __builtin_amdgcn_global_load_async_to_lds_b128(
      (gi32x4*)(uintptr_t)g,
      (li32x4*)(uintptr_t)loff,
      0, 0);
#else
  asm volatile("global_load_async_to_lds_b128 %0, %1, off"
               :: "v"(loff), "v"(g) : "memory");
#endif
}

__device__ __forceinline__ void wait_async_zero() {
#if __has_builtin(__builtin_amdgcn_s_wait_asynccnt)
  __builtin_amdgcn_s_wait_asynccnt(0);
#else
  asm volatile("s_wait_asynccnt 0" ::: "memory");
#endif
}

// ---------------- kernel 0: x0 = (pred-target)^2, init control ----------------
__global__ void hdl_init_sq(const float* __restrict__ pred, const float* __restrict__ targ,
                            float* __restrict__ bufA,
                            float* scaleArr, float* biasArr, double* totalAccum) {
  size_t i = (size_t)blockIdx.x * blockDim.x + threadIdx.x;   // over N_TOTAL/4 vec4's
  const v4f* p4 = (const v4f*)pred;
  const v4f* t4 = (const v4f*)targ;
  v4f p = __builtin_nontemporal_load(p4 + i);   // NT: keep L2 free for ping-pong bufs
  v4f t = __builtin_nontemporal_load(t4 + i);
  v4f d = p - t;
  ((v4f*)bufA)[i] = d * d;
  if (blockIdx.x == 0) {
    if (threadIdx.x < 8) { scaleArr[threadIdx.x] = 1.0f; biasArr[threadIdx.x] = 0.0f; }
    if (threadIdx.x == 8) { *totalAccum = 0.0; }
  }
}

// ---------------- stencil + per-block reduction ----------------
// One block = 64 Y-rows (half of a (b,x) slice). Tile in LDS with zero halo.
// Interior column starts at LDS column 4 so async B128 LDS writes are 16B aligned.
// 5-point cross computed on the matrix pipe: T'X + XT' with T' = tridiag(1, 0.5, 1).
#define TROWS 66
#define TPITCH 136   // floats per tile row (128 interior + halos + pad)

__global__ void hdl_stencil(const float* __restrict__ src, float* __restrict__ dst,
                            const float* __restrict__ scaleArr,
                            const float* __restrict__ biasArr,
                            float* __restrict__ pmin, float* __restrict__ pmax,
                            double* __restrict__ psum) {
  __shared__ __align__(16) float tile[TROWS * TPITCH];  // 35,904 B
  __shared__ float rmin[8], rmax[8];
  __shared__ double rsum[8];

  const int blk  = blockIdx.x;        // 0..2047
  const int s    = blk >> 1;          // slice id 0..1023  (= b*128 + x)
  const int half = blk & 1;
  const int b    = s >> 7;
  const int t    = threadIdx.x;       // 256 threads = 8 wave32
  const int wave = t >> 5, lane = t & 31;
  const int y0   = half * 64;

  // zero side halo columns (cols 3 and 132 of each tile row)
  if (t < TROWS) {
    tile[t * TPITCH + 3]   = 0.0f;
    tile[t * TPITCH + 132] = 0.0f;
  }

  // fill rows: tile row r <-> global row gy = y0-1+r ; 32 x 16B chunks per row
  const float* sliceBase = src + (size_t)s * SLICE_EL;
  const v4f z4 = {0.f, 0.f, 0.f, 0.f};
  for (int i = 0; i < 9; ++i) {
    int c = i * 256 + t;              // 0 .. 2111 (66 rows * 32 chunks)
    if (c < TROWS * 32) {
      int r   = c >> 5;
      int col = c & 31;
      float* lp = &tile[r * TPITCH + 4 + col * 4];  // 16B aligned
      int gy = y0 - 1 + r;
      if (gy >= 0 && gy < 128) {
        async_load_b128(sliceBase + gy * 128 + col * 4, lp);
      } else {
        *(v4f*)lp = z4;               // zero padding rows
      }
    }
  }
  wait_async_zero();
  __syncthreads();

  const float sc = scaleArr[b];
  const float bi = biasArr[b];
  const float inv7 = 1.0f / 7.0f;

  float lmin = 3.402823466e38f, lmax = -3.402823466e38f, lsum = 0.0f;

#if __has_builtin(__builtin_amdgcn_wmma_f32_16x16x4_f32)
  // ---- WMMA path: per wave, one 16-col strip = 4 tiles of 16x16 ----
  const int lh = lane >> 4;           // lane half
  const int mn = lane & 15;           // m (A-frag) / n (B-frag) index
  // T' fragment: A[m][k] layout == B[k][n] layout => one fragment serves both.
  v2f tf[4];
#pragma unroll
  for (int c = 0; c < 4; ++c) {
#pragma unroll
    for (int v = 0; v < 2; ++v) {
      int kg = 4 * c + 2 * lh + v;
      int d  = mn - kg;
      tf[c][v] = (d == 1 || d == -1) ? 1.0f : (d == 0 ? 0.5f : 0.0f);
    }
  }
  const int tz = wave;                // tile column 0..7
  const int Cz = tz * 16 + 4;         // LDS column of tile col 0
#pragma unroll
  for (int ty = 0; ty < 4; ++ty) {
    const int Rt = ty * 16 + 1;       // LDS row of tile row 0
    v8f acc = {0.f, 0.f, 0.f, 0.f, 0.f, 0.f, 0.f, 0.f};
#pragma unroll
    for (int c = 0; c < 4; ++c) {
      const int kl0 = 4 * c + 2 * lh;
      v2f xrow, xcol;
      xrow.x = tile[(Rt + kl0    ) * TPITCH + Cz + mn];   // B-frag of T'·X
      xrow.y = tile[(Rt + kl0 + 1) * TPITCH + Cz + mn];
      xcol.x = tile[(Rt + mn) * TPITCH + Cz + kl0    ];   // A-frag of X·T'
      xcol.y = tile[(Rt + mn) * TPITCH + Cz + kl0 + 1];
      acc = __builtin_amdgcn_wmma_f32_16x16x4_f32(
                false, tf[c], false, xrow, (short)0, acc, false, false);
      acc = __builtin_amdgcn_wmma_f32_16x16x4_f32(
                false, xcol, false, tf[c], (short)0, acc, false, false);
    }
    // epilogue: halo taps, affine+relu, store, local reduce
#pragma unroll
    for (int r = 0; r < 8; ++r) {
      const int m  = r + 8 * lh;      // C/D layout: VGPR r, lane half -> row
      const int n  = mn;
      const int gy = y0 + ty * 16 + m;
      const int z  = tz * 16 + n;
      float val = acc[r];
      if (r == 0 && lh == 0) val += tile[(Rt - 1 ) * TPITCH + Cz + n];  // m==0
      if (r == 7 && lh == 1) val += tile[(Rt + 16) * TPITCH + Cz + n];  // m==15
      if (n == 0 )           val += tile[(Rt + m) * TPITCH + Cz - 1 ];
      if (n == 15)           val += tile[(Rt + m) * TPITCH + Cz + 16];
      int cnt = 5 - (gy == 0) - (gy == 127) - (z == 0) - (z == 127);
      float vv = (sc * val + (float)cnt * bi) * inv7 - 0.5f;
      float e = vv > 0.0f ? vv : 0.0f;
      dst[(size_t)s * SLICE_EL + (size_t)gy * 128 + z] = e;
      lmin = fminf(lmin, e);
      lmax = fmaxf(lmax, e);
      lsum += e;
    }
  }
#else
  // ---- scalar fallback: classic 5-point gather ----
  for (int i = 0; i < 32; ++i) {
    int idx = i * 256 + t;
    int ly  = idx >> 7;
    int z   = idx & 127;
    int gy  = y0 + ly;
    const float* cr = &tile[(ly + 1) * TPITCH + (z + 4)];
    float raw = cr[0] + cr[1] + cr[-1] + cr[TPITCH] + cr[-TPITCH];
    int cnt = 5 - (gy == 0) - (gy == 127) - (z == 0) - (z == 127);
    float v = (sc * raw + (float)cnt * bi) * inv7 - 0.5f;
    float e = v > 0.0f ? v : 0.0f;
    dst[(size_t)s * SLICE_EL + (size_t)gy * 128 + z] = e;
    lmin = fminf(lmin, e);
    lmax = fmaxf(lmax, e);
    lsum += e;
  }
#endif

  // wave32 tree reduce
  for (int m = 16; m > 0; m >>= 1) {
    lmin = fminf(lmin, __shfl_xor(lmin, m, 32));
    lmax = fmaxf(lmax, __shfl_xor(lmax, m, 32));
    lsum += __shfl_xor(lsum, m, 32);
  }
  if (lane == 0) { rmin[wave] = lmin; rmax[wave] = lmax; rsum[wave] = (double)lsum; }
  __syncthreads();
  if (t == 0) {
    float  m0 = rmin[0], M0 = rmax[0];
    double S0 = rsum[0];
    for (int w = 1; w < 8; ++w) {
      m0 = fminf(m0, rmin[w]); M0 = fmaxf(M0, rmax[w]); S0 += rsum[w];
    }
    pmin[blk] = m0; pmax[blk] = M0; psum[blk] = S0;
  }
}

// ---------------- finalize: per-batch ptp, scale/bias, weighted sum ----------------
__global__ void hdl_finalize(const float* __restrict__ pmin, const float* __restrict__ pmax,
                             const double* __restrict__ psum,
                             float* scaleArr, float* biasArr, double* totalAccum,
                             float wk, int last, float* out) {
  __shared__ double cb[8];
  const int t = threadIdx.x;          // 256 threads; wave w handles batch w
  const int w = t >> 5, lane = t & 31;

  float  m = 3.402823466e38f, M = -3.402823466e38f;
  double S = 0.0;
  for (int j = 0; j < 8; ++j) {       // 256 partial blocks per batch
    int p = w * 256 + lane * 8 + j;
    m = fminf(m, pmin[p]);
    M = fmaxf(M, pmax[p]);
    S += psum[p];
  }
  for (int sft = 16; sft > 0; sft >>= 1) {
    m = fminf(m, __shfl_xor(m, sft, 32));
    M = fmaxf(M, __shfl_xor(M, sft, 32));
    S += __shfl_xor(S, sft, 32);
  }
  if (lane == 0) {
    float ptp = M - m;
    float scv, biv; double contrib;
    if (ptp != 0.0f) {
      scv = 1.0f / ptp;
      biv = -m * scv;
      contrib = (S - (double)N_BATCH * (double)m) / (double)ptp;
    } else {
      scv = 1.0f; biv = 0.0f; contrib = S;
    }
    scaleArr[w] = scv; biasArr[w] = biv; cb[w] = contrib;
  }
  __syncthreads();
  if (t == 0) {
    double acc = *totalAccum;
    for (int i = 0; i < 8; ++i) acc += (double)wk * cb[i];
    *totalAccum = acc;
    if (last) *out = (float)(acc / (double)N_TOTAL);
  }
}

// ---------------- host entry ----------------
extern "C" void kernel_launch(void* const* d_in, const int* in_sizes, int n_in,
                              void* d_out, int out_size, void* d_ws, size_t ws_size,
                              hipStream_t stream) {
  (void)in_sizes; (void)n_in; (void)out_size; (void)ws_size;
  const float* pred = (const float*)d_in[0];
  const float* targ = (const float*)d_in[1];
  char* ws = (char*)d_ws;

  double* totalAccum = (double*)(ws + WS_TOTAL_OFF);
  float*  scaleArr   = (float*) (ws + WS_SCALE_OFF);
  float*  biasArr    = (float*) (ws + WS_BIAS_OFF);
  float*  pmin       = (float*) (ws + WS_PMIN_OFF);
  float*  pmax       = (float*) (ws + WS_PMAX_OFF);
  double* psum       = (double*)(ws + WS_PSUM_OFF);
  float*  bufA       = (float*) (ws + WS_BUFA_OFF);
  float*  bufB       = (float*) (ws + WS_BUFB_OFF);

  hdl_init_sq<<<N_TOTAL / 4 / 256, 256, 0, stream>>>(pred, targ, bufA,
                                                     scaleArr, biasArr, totalAccum);

  const float* src = bufA;
  float* dst = bufB;
  for (int k = 0; k < 10; ++k) {
    hdl_stencil<<<NBLK, 256, 0, stream>>>(src, dst, scaleArr, biasArr, pmin, pmax, psum);
    float wk = (float)((k + 1) * (k + 1));   // ALPHA = 2.0
    hdl_finalize<<<1, 256, 0, stream>>>(pmin, pmax, psum, scaleArr, biasArr,
                                        totalAccum, wk, (k == 9) ? 1 : 0, (float*)d_out);
    const float* tswap = dst; dst = (float*)src; src = tswap;
  }
}